// GenTargets_74766790689175
// MI455X (gfx1250) — compile-verified
//
#include <hip/hip_runtime.h>
#include <math.h>

// FCOS target generation for MI455X (gfx1250, wave32).
//
// B=8 batches, M=64 GT boxes, HW=17064 locations across 5 FPN levels
// (100x128, 50x64, 25x32, 13x16, 7x8; strides 8..128).
// Output (flat f32, reference tuple order):
//   out[0        .. B*HW)    cls (label value as float)
//   out[B*HW     .. 2*B*HW)  centerness
//   out[2*B*HW   .. 6*B*HW)  reg (l,t,r,b)
//
// CDNA5 usage: async global->LDS broadcast of the per-batch box/label table
// (global_load_async_to_lds_b128/b32 + s_wait_asynccnt), global_prefetch_b8,
// ds_load_b128 inner loop. WMMA deliberately unused: masked argmin over
// compares/selects has no matrix-multiply structure.

#define AS1 __attribute__((address_space(1)))
#define AS3 __attribute__((address_space(3)))

typedef __attribute__((ext_vector_type(4))) int v4i;

#ifndef __has_builtin
#define __has_builtin(x) 0
#endif

#if __has_builtin(__builtin_amdgcn_global_load_async_to_lds_b128) && \
    __has_builtin(__builtin_amdgcn_global_load_async_to_lds_b32)  && \
    __has_builtin(__builtin_amdgcn_s_wait_asynccnt)
#define FCOS_HAVE_ASYNC_LDS 1
#else
#define FCOS_HAVE_ASYNC_LDS 0
#endif

static constexpr int kB  = 8;
static constexpr int kM  = 64;
static constexpr int kHW = 17064;          // 12800 + 3200 + 800 + 208 + 56
static constexpr float kBIG = 99999999.0f;

__global__ __launch_bounds__(256) void fcos_targets_kernel(
    const float* __restrict__ gt_box,   // (B, M, 4)
    const int*   __restrict__ labels,   // (B, M)
    float*       __restrict__ out)      // cls | center | reg, flat f32
{
  __shared__ float4 sbox[kM];
  __shared__ int    slab[kM];

  const int b = blockIdx.y;
  const int t = threadIdx.x;

  const float* gb = gt_box + (size_t)b * kM * 4;
  const int*   lb = labels + (size_t)b * kM;

  // Warm the (tiny, shared-by-all-blocks) tables: emits global_prefetch_b8.
  __builtin_prefetch((const void*)gb, 0, 3);
  __builtin_prefetch((const void*)lb, 0, 3);

#if FCOS_HAVE_ASYNC_LDS
  // gfx1250 async copy path: 64 lanes (waves 0-1) each move one float4 box
  // and one int label straight into LDS, tracked by ASYNCcnt.
  if (t < kM) {
    __builtin_amdgcn_global_load_async_to_lds_b128(
        (AS1 v4i*)(AS1 void*)(gb + t * 4),
        (AS3 v4i*)(AS3 void*)&sbox[t], 0, 0);
    __builtin_amdgcn_global_load_async_to_lds_b32(
        (AS1 int*)(AS1 void*)(lb + t),
        (AS3 int*)(AS3 void*)&slab[t], 0, 0);
  }
  __builtin_amdgcn_s_wait_asynccnt(0);
  __syncthreads();
#else
  if (t < kM) {
    sbox[t] = reinterpret_cast<const float4*>(gb)[t];
    slab[t] = lb[t];
  }
  __syncthreads();
#endif

  const int p = blockIdx.x * 256 + t;
  if (p >= kHW) return;

  // Locate FPN level from the flat location index.
  int lvl, local;
  if      (p < 12800) { lvl = 0; local = p;         }
  else if (p < 16000) { lvl = 1; local = p - 12800; }
  else if (p < 16800) { lvl = 2; local = p - 16000; }
  else if (p < 17008) { lvl = 3; local = p - 16800; }
  else                { lvl = 4; local = p - 17008; }

  const int   stride = 8 << lvl;                 // 8,16,32,64,128
  const int   wmask  = (128 >> lvl) - 1;         // widths 128,64,32,16,8 (pow2)
  const int   ysh    = 7 - lvl;
  const float lo  = (lvl == 0) ? -1.0f : (float)(32 << lvl);  // -1,64,128,256,512
  const float hi  = (lvl == 4) ? kBIG  : (float)(64 << lvl);  // 64,128,256,512,BIG
  const float rad = (float)stride * 1.5f;

  const float x = (float)((local & wmask) * stride + (stride >> 1));
  const float y = (float)((local >> ysh)  * stride + (stride >> 1));

  // Masked argmin over 64 boxes; strict '<' reproduces jnp.argmin first-min.
  float bestArea = kBIG;
  int   best = 0;
  bool  pos  = false;

  #pragma unroll 8
  for (int m = 0; m < kM; ++m) {
    const float4 g  = sbox[m];
    const float l   = x - g.x;
    const float tt  = y - g.y;
    const float r   = g.z - x;
    const float bb  = g.w - y;
    const float omin = fminf(fminf(l, tt), fminf(r, bb));
    const float omax = fmaxf(fmaxf(l, tt), fmaxf(r, bb));
    const float area = (l + r) * (tt + bb);
    const float cx   = (g.x + g.z) * 0.5f;
    const float cy   = (g.y + g.w) * 0.5f;
    const float gmax = fmaxf(fabsf(x - cx), fabsf(y - cy));
    const bool ok = (omin > 0.0f) & (omax > lo) & (omax <= hi) & (gmax < rad);
    const float a = ok ? area : kBIG;
    if (a < bestArea) { bestArea = a; best = m; }
    pos = pos | ok;
  }

  // Recompute offsets of the winning box once.
  const float4 g = sbox[best];
  const float l  = x - g.x;
  const float tt = y - g.y;
  const float r  = g.z - x;
  const float bb = g.w - y;

  const float lrmin = fminf(l, r),   lrmax = fmaxf(l, r);
  const float tbmin = fminf(tt, bb), tbmax = fmaxf(tt, bb);
  const float ratio = (lrmin * tbmin) / (lrmax * tbmax + 1e-10f);

  const float centerv = pos ? sqrtf(ratio) : -1.0f;
  const float clsv    = pos ? (float)slab[best] : 0.0f;

  const size_t o = (size_t)b * kHW + (size_t)p;
  out[o]                    = clsv;
  out[(size_t)kB * kHW + o] = centerv;
  float* reg = out + (size_t)2 * kB * kHW + o * 4;
  reg[0] = pos ? l  : -1.0f;
  reg[1] = pos ? tt : -1.0f;
  reg[2] = pos ? r  : -1.0f;
  reg[3] = pos ? bb : -1.0f;
}

extern "C" void kernel_launch(void* const* d_in, const int* in_sizes, int n_in,
                              void* d_out, int out_size, void* d_ws, size_t ws_size,
                              hipStream_t stream) {
  (void)in_sizes; (void)n_in; (void)out_size; (void)d_ws; (void)ws_size;
  const float* gt_box = (const float*)d_in[0];   // (8, 64, 4) f32
  const int*   labels = (const int*)d_in[1];     // (8, 64)    i32
  float*       out    = (float*)d_out;           // 6*8*17064 f32

  dim3 grid((kHW + 255) / 256, kB, 1);           // 67 x 8 blocks
  dim3 block(256, 1, 1);                         // 8 wave32 waves
  hipLaunchKernelGGL(fcos_targets_kernel, grid, block, 0, stream,
                     gt_box, labels, out);
}